// LinearAttention_81509889343716
// MI455X (gfx1250) — compile-verified
//
#include <hip/hip_runtime.h>
#include <math.h>

// ---------------------------------------------------------------------------
// LinearAttention forward for MI455X (gfx1250), fp32 end-to-end.
// Matrix math uses V_WMMA_F32_16X16X4_F32 (wave32, 16x16 tiles, K=4).
//
// LDS layouts are chosen for the WMMA lane maps:
//   A frag (16x4): lanes 0-15 row M=lane K=kk,kk+1; lanes 16-31 K=kk+2,kk+3.
//   B frag (4x16): lanes 0-15 col N=lane K=kk,kk+1; lanes 16-31 K=kk+2,kk+3.
//   C/D: vgpr r holds M = r + 8*(lane>=16), N = lane%16.
// So B wants K-contiguous pairs per (N) column -> store tiles K-major
// (transposed), pitch 132 floats (== 4 mod 64 banks -> conflict-free b64).
// ---------------------------------------------------------------------------

#define HEADS 4
#define DH    32
#define CDIM  128          // dim == hidden == 128
#define NSEQ  16384        // H*W
#define BATCH 8
#define NT    64           // n-tile, k_qkv
#define NTO   128          // n-tile, k_out
#define XPITCH 132         // transposed-tile pitch (floats), 4 mod 64 banks
#define KVPITCH 68         // kv row pitch (floats), 4 mod 64 banks

typedef __attribute__((ext_vector_type(2))) float v2f;
typedef __attribute__((ext_vector_type(8))) float v8f;

__device__ __forceinline__ v8f wmma4(v2f a, v2f b, v8f c) {
  return __builtin_amdgcn_wmma_f32_16x16x4_f32(false, a, false, b,
                                               (short)0, c, false, false);
}

// ---------------------------------------------------------------------------
// Kernel 1: qkv GEMM (384x128 @ 128xNT) with q/k/v sharing B fragments,
// fused k-softmax(d) and ctx[h] += k_softT * v; q stored raw.
// ---------------------------------------------------------------------------
__global__ __launch_bounds__(256)
void k_qkv(const float* __restrict__ x, const float* __restrict__ w_qkv,
           float* __restrict__ q_buf, float* __restrict__ ctx) {
  extern __shared__ float smem[];
  float* xs = smem;                    // transposed x tile [NT][XPITCH]
  float* kv = smem + NT * XPITCH;      // [256][KVPITCH] k rows 0-127, v 128-255

  const int t = threadIdx.x;
  const int lane = t & 31, wave = t >> 5;
  const int lo = lane & 15, hi = lane >> 4;
  const int n0 = blockIdx.x * NT;
  const int b  = blockIdx.y;

  // ---- stage x tile transposed: coalesced float4 reads, scatter b32 stores
  const float* xb = x + ((size_t)b * CDIM) * NSEQ + n0;
#pragma unroll
  for (int i = 0; i < 8; ++i) {
    int f  = t + 256 * i;              // 2048 float4s
    int c  = f >> 4;                   // channel row 0..127
    int n4 = (f & 15) << 2;            // n offset 0..60
    float4 v = *(const float4*)(xb + (size_t)c * NSEQ + n4);
    xs[(n4 + 0) * XPITCH + c] = v.x;
    xs[(n4 + 1) * XPITCH + c] = v.y;
    xs[(n4 + 2) * XPITCH + c] = v.z;
    xs[(n4 + 3) * XPITCH + c] = v.w;
  }
  __syncthreads();

  // ---- WMMA GEMM: one 16-row strip each of q/k/v per wave, 4 col tiles.
  // Per K-step: 3 A loads + 4 B loads feed 12 WMMAs (B reused 3x).
  const float* wr = w_qkv + (size_t)(wave * 16 + lo) * CDIM + 2 * hi;
  __builtin_prefetch(wr, 0, 0);
  __builtin_prefetch(wr + 128 * CDIM, 0, 0);
  __builtin_prefetch(wr + 256 * CDIM, 0, 0);
  v8f acc[3][4] = {};
  for (int kk = 0; kk < CDIM; kk += 4) {
    v2f a0 = *(const v2f*)(wr + kk);
    v2f a1 = *(const v2f*)(wr + 128 * CDIM + kk);
    v2f a2 = *(const v2f*)(wr + 256 * CDIM + kk);
#pragma unroll
    for (int ct = 0; ct < 4; ++ct) {
      v2f bf = *(const v2f*)(xs + (ct * 16 + lo) * XPITCH + kk + 2 * hi);
      acc[0][ct] = wmma4(a0, bf, acc[0][ct]);
      acc[1][ct] = wmma4(a1, bf, acc[1][ct]);
      acc[2][ct] = wmma4(a2, bf, acc[2][ct]);
    }
  }
  // q strip -> global
  {
    float* qb = q_buf + ((size_t)b * CDIM + wave * 16) * NSEQ + n0;
#pragma unroll
    for (int ct = 0; ct < 4; ++ct)
#pragma unroll
      for (int r = 0; r < 8; ++r)
        qb[(size_t)(r + 8 * hi) * NSEQ + ct * 16 + lo] = acc[0][ct][r];
  }
  // k, v strips -> LDS (k at rows 0-127, v at rows 128-255)
#pragma unroll
  for (int s = 1; s < 3; ++s) {
    float* kvb = kv + ((s - 1) * CDIM + wave * 16) * KVPITCH;
#pragma unroll
    for (int ct = 0; ct < 4; ++ct)
#pragma unroll
      for (int r = 0; r < 8; ++r)
        kvb[(r + 8 * hi) * KVPITCH + ct * 16 + lo] = acc[s][ct][r];
  }
  __syncthreads();

  // ---- k softmax over d (32 per head) per (head, column) ----
  {
    const int h = t >> 6, j = t & 63;                // 4 heads x 64 cols
    float* kc = kv + (h * DH) * KVPITCH + j;
    float m = -3.4e38f;
#pragma unroll
    for (int d = 0; d < DH; ++d) m = fmaxf(m, kc[d * KVPITCH]);
    float s = 0.f;
#pragma unroll
    for (int d = 0; d < DH; ++d) {
      float e = __expf(kc[d * KVPITCH] - m); kc[d * KVPITCH] = e; s += e;
    }
    const float is = 1.f / s;
#pragma unroll
    for (int d = 0; d < DH; ++d) kc[d * KVPITCH] *= is;
  }
  __syncthreads();

  // ---- ctx[h][d][e] += sum_n ksoft[d][n] * v[e][n] : WMMA, K = n-tile ----
  {
    const int h  = wave >> 1;                        // 2 waves per head
    const int d0 = (wave & 1) * 16;
    const float* ks = kv + (h * DH) * KVPITCH;
    const float* vs = kv + (CDIM + h * DH) * KVPITCH;
    v8f cacc[2] = {};
    for (int kk = 0; kk < NT; kk += 4) {
      v2f a = *(const v2f*)(ks + (d0 + lo) * KVPITCH + kk + 2 * hi);
#pragma unroll
      for (int et = 0; et < 2; ++et) {
        v2f bf = *(const v2f*)(vs + (et * 16 + lo) * KVPITCH + kk + 2 * hi);
        cacc[et] = wmma4(a, bf, cacc[et]);
      }
    }
    float* cb = ctx + ((size_t)b * HEADS + h) * DH * DH;
#pragma unroll
    for (int et = 0; et < 2; ++et)
#pragma unroll
      for (int r = 0; r < 8; ++r)
        atomicAdd(cb + (d0 + r + 8 * hi) * DH + et * 16 + lo, cacc[et][r]);
  }
}

// ---------------------------------------------------------------------------
// Kernel 2: per-(b, channel) softmax stats of q over n (max, sum of exp).
// ---------------------------------------------------------------------------
__global__ __launch_bounds__(256)
void k_qstats(const float* __restrict__ q_buf,
              float* __restrict__ qmax, float* __restrict__ qsum) {
  __shared__ float red[256];
  const int c = blockIdx.x, b = blockIdx.y, t = threadIdx.x;
  const float4* row4 = (const float4*)(q_buf + ((size_t)b * CDIM + c) * NSEQ);
  float m = -3.4e38f;
  for (int i = t; i < NSEQ / 4; i += 256) {
    float4 v = row4[i];
    m = fmaxf(m, fmaxf(fmaxf(v.x, v.y), fmaxf(v.z, v.w)));
  }
  red[t] = m; __syncthreads();
  for (int s = 128; s > 0; s >>= 1) {
    if (t < s) red[t] = fmaxf(red[t], red[t + s]);
    __syncthreads();
  }
  m = red[0]; __syncthreads();
  float sum = 0.f;
  for (int i = t; i < NSEQ / 4; i += 256) {
    float4 v = row4[i];
    sum += __expf(v.x - m) + __expf(v.y - m) + __expf(v.z - m) + __expf(v.w - m);
  }
  red[t] = sum; __syncthreads();
  for (int s = 128; s > 0; s >>= 1) {
    if (t < s) red[t] += red[t + s];
    __syncthreads();
  }
  if (t == 0) { qmax[b * CDIM + c] = m; qsum[b * CDIM + c] = red[0]; }
}

// ---------------------------------------------------------------------------
// Kernel 3: fold W_out and ctx:
// W2[c][h*32+d] = scale * sum_e W_out[c][h*32+e] * ctx[h][d][e]
// ---------------------------------------------------------------------------
__global__ __launch_bounds__(256)
void k_w2(const float* __restrict__ w_out, const float* __restrict__ ctx,
          float* __restrict__ W2) {
  const int b = blockIdx.x, t = threadIdx.x;
  const float scale = 0.17677669529663687f;   // DIM_HEAD ** -0.5
  for (int idx = t; idx < CDIM * CDIM; idx += 256) {
    const int c = idx >> 7, o = idx & 127;
    const int h = o >> 5,  d = o & 31;
    const float* wrp = w_out + (size_t)c * CDIM + h * DH;
    const float* crp = ctx + (((size_t)b * HEADS + h) * DH + d) * DH;
    float s = 0.f;
#pragma unroll
    for (int e = 0; e < DH; ++e) s += wrp[e] * crp[e];
    W2[((size_t)b * CDIM + c) * CDIM + o] = s * scale;
  }
}

// ---------------------------------------------------------------------------
// Kernel 4: y = W2 @ softmax_n(q) + b_out (WMMA), fused GroupNorm stats.
// ---------------------------------------------------------------------------
__global__ __launch_bounds__(256)
void k_out(const float* __restrict__ q_buf, const float* __restrict__ W2,
           const float* __restrict__ b_out, const float* __restrict__ qmax,
           const float* __restrict__ qsum, float* __restrict__ y_buf,
           float* __restrict__ gnsum, float* __restrict__ gnsq) {
  extern __shared__ float smem[];
  float* qs    = smem;                      // transposed q tile [NTO][XPITCH]
  float* sm_m  = smem + NTO * XPITCH;       // [128] per-channel max
  float* sm_is = sm_m + CDIM;               // [128] per-channel 1/sum
  float* red   = sm_is + CDIM;              // [512] reduction scratch
  const int t = threadIdx.x, lane = t & 31, wave = t >> 5;
  const int lo = lane & 15, hi = lane >> 4;
  const int n0 = blockIdx.x * NTO, b = blockIdx.y;

  if (t < CDIM) {
    sm_m[t]  = qmax[b * CDIM + t];
    sm_is[t] = 1.f / qsum[b * CDIM + t];
  }
  __syncthreads();

  // stage softmax-normalized q tile, transposed
  const float* qb = q_buf + ((size_t)b * CDIM) * NSEQ + n0;
#pragma unroll
  for (int i = 0; i < 16; ++i) {
    int f  = t + 256 * i;                   // 4096 float4s
    int c  = f >> 5;                        // channel 0..127
    int n4 = (f & 31) << 2;                 // n offset 0..124
    float4 v = *(const float4*)(qb + (size_t)c * NSEQ + n4);
    const float m = sm_m[c], is = sm_is[c];
    qs[(n4 + 0) * XPITCH + c] = __expf(v.x - m) * is;
    qs[(n4 + 1) * XPITCH + c] = __expf(v.y - m) * is;
    qs[(n4 + 2) * XPITCH + c] = __expf(v.z - m) * is;
    qs[(n4 + 3) * XPITCH + c] = __expf(v.w - m) * is;
  }
  __syncthreads();

  // WMMA: one 16-row strip per wave, 8 col tiles; 1 A + 8 B loads -> 8 WMMAs
  const int oc0 = wave * 16;
  const float* wrp = W2 + ((size_t)b * CDIM + oc0 + lo) * CDIM + 2 * hi;
  v8f acc[8] = {};
  for (int kk = 0; kk < CDIM; kk += 4) {
    v2f a = *(const v2f*)(wrp + kk);
#pragma unroll
    for (int ct = 0; ct < 8; ++ct) {
      v2f bf = *(const v2f*)(qs + (ct * 16 + lo) * XPITCH + kk + 2 * hi);
      acc[ct] = wmma4(a, bf, acc[ct]);
    }
  }

  // bias, store y, accumulate GroupNorm sum / sumsq
  float ls = 0.f, lq = 0.f;
  float* yb = y_buf + ((size_t)b * CDIM + oc0) * NSEQ + n0;
#pragma unroll
  for (int r = 0; r < 8; ++r) {
    const float bias = b_out[oc0 + r + 8 * hi];
#pragma unroll
    for (int ct = 0; ct < 8; ++ct) {
      const float v = acc[ct][r] + bias;
      yb[(size_t)(r + 8 * hi) * NSEQ + ct * 16 + lo] = v;
      ls += v; lq += v * v;
    }
  }
  __syncthreads();
  red[t] = ls; red[256 + t] = lq;
  __syncthreads();
  for (int s = 128; s > 0; s >>= 1) {
    if (t < s) { red[t] += red[t + s]; red[256 + t] += red[256 + t + s]; }
    __syncthreads();
  }
  if (t == 0) { atomicAdd(&gnsum[b], red[0]); atomicAdd(&gnsq[b], red[256]); }
}

// ---------------------------------------------------------------------------
// Kernel 5: GroupNorm(1, C) normalize + affine, write final output.
// ---------------------------------------------------------------------------
__global__ __launch_bounds__(256)
void k_gn(const float* __restrict__ y_buf, const float* __restrict__ gnsum,
          const float* __restrict__ gnsq, const float* __restrict__ gn_w,
          const float* __restrict__ gn_b, float* __restrict__ out) {
  const size_t gid  = (size_t)blockIdx.x * 256 + threadIdx.x;  // float4 index
  const size_t flat = gid * 4;
  const int b = (int)(flat / ((size_t)CDIM * NSEQ));
  const int c = (int)((flat % ((size_t)CDIM * NSEQ)) / NSEQ);
  const float inv_n = 1.f / ((float)CDIM * (float)NSEQ);
  const float mu  = gnsum[b] * inv_n;
  const float var = gnsq[b] * inv_n - mu * mu;
  const float inv = rsqrtf(var + 1e-5f);
  const float g = gn_w[c] * inv, bb = gn_b[c];
  float4 v = *(const float4*)(y_buf + flat);
  v.x = (v.x - mu) * g + bb; v.y = (v.y - mu) * g + bb;
  v.z = (v.z - mu) * g + bb; v.w = (v.w - mu) * g + bb;
  *(float4*)(out + flat) = v;
}

// ---------------------------------------------------------------------------
extern "C" void kernel_launch(void* const* d_in, const int* in_sizes, int n_in,
                              void* d_out, int out_size, void* d_ws, size_t ws_size,
                              hipStream_t stream) {
  const float* x     = (const float*)d_in[0];
  const float* w_qkv = (const float*)d_in[1];
  const float* w_out = (const float*)d_in[2];
  const float* b_out = (const float*)d_in[3];
  const float* gn_w  = (const float*)d_in[4];
  const float* gn_b  = (const float*)d_in[5];

  float* ws    = (float*)d_ws;
  float* q_buf = ws;                                    // 8*128*16384 (64 MB)
  float* y_buf = q_buf + (size_t)BATCH * CDIM * NSEQ;   // 64 MB
  float* ctx   = y_buf + (size_t)BATCH * CDIM * NSEQ;   // 8*4*32*32
  float* gnsum = ctx + BATCH * HEADS * DH * DH;         // 8
  float* gnsq  = gnsum + BATCH;                         // 8
  float* qmax  = gnsq + BATCH;                          // 1024
  float* qsum  = qmax + BATCH * CDIM;                   // 1024
  float* W2    = qsum + BATCH * CDIM;                   // 8*128*128

  // zero the atomically-accumulated buffers (graph-capture legal)
  hipMemsetAsync(ctx, 0,
                 (size_t)(BATCH * HEADS * DH * DH + 2 * BATCH) * sizeof(float),
                 stream);

  const size_t lds1 = (size_t)(NT * XPITCH + 256 * KVPITCH) * sizeof(float);
  k_qkv<<<dim3(NSEQ / NT, BATCH), 256, lds1, stream>>>(x, w_qkv, q_buf, ctx);

  k_qstats<<<dim3(CDIM, BATCH), 256, 0, stream>>>(q_buf, qmax, qsum);
  k_w2<<<BATCH, 256, 0, stream>>>(w_out, ctx, W2);

  const size_t lds4 = (size_t)(NTO * XPITCH + 2 * CDIM + 512) * sizeof(float);
  k_out<<<dim3(NSEQ / NTO, BATCH), 256, lds4, stream>>>(
      q_buf, W2, b_out, qmax, qsum, y_buf, gnsum, gnsq);

  const size_t n_f4 = (size_t)BATCH * CDIM * NSEQ / 4;
  k_gn<<<(unsigned)(n_f4 / 256), 256, 0, stream>>>(y_buf, gnsum, gnsq, gn_w, gn_b,
                                                   (float*)d_out);
}